// CascadedLoRALinear4bit_10479720202377
// MI455X (gfx1250) — compile-verified
//
#include <hip/hip_runtime.h>
#include <hip/hip_bf16.h>

typedef __attribute__((ext_vector_type(2))) float v2f;
typedef __attribute__((ext_vector_type(8))) float v8f;
typedef __attribute__((ext_vector_type(4))) int   v4i;

#define M_TOT 8192
#define K_DIM 4096
#define N_DIM 4096
#define R1 64
#define R2 32
#define S1 4.0f

// ---- CDNA5 async global->LDS path (guarded; falls back to VGPR staging) ----
#if defined(__has_builtin)
#if __has_builtin(__builtin_amdgcn_global_load_async_to_lds_b128) && \
    __has_builtin(__builtin_amdgcn_s_wait_asynccnt)
#define USE_ASYNC 1
#endif
#endif
#ifndef USE_ASYNC
#define USE_ASYNC 0
#endif

#if USE_ASYNC
__device__ __forceinline__ void async_copy16(const float* g, float* l) {
    __builtin_amdgcn_global_load_async_to_lds_b128((v4i*)g, (v4i*)l, 0, 0);
}
#endif

// ---------------------------------------------------------------------------
// Kernel 1: h1 = S1 * (x @ A^T)  [M,64]   and   u = x @ A1^T  [M,32]
// ---------------------------------------------------------------------------
__global__ __launch_bounds__(256)
void lora_proj_kernel(const float* __restrict__ x,
                      const float* __restrict__ A,
                      const float* __restrict__ A1,
                      float* __restrict__ h1,
                      float* __restrict__ u)
{
    __shared__ float Xs[128][36];
    __shared__ float As[96][36];

    const int tid   = threadIdx.x;
    const int wave  = tid >> 5;
    const int lane  = tid & 31;
    const int l16   = lane & 15;
    const int koff  = (lane >> 4) * 2;
    const int rBase = wave * 16;
    const int rowBase = blockIdx.x * 128;

    v8f acc[6];
#pragma unroll
    for (int j = 0; j < 6; ++j) acc[j] = (v8f){0,0,0,0,0,0,0,0};

    for (int ck = 0; ck < K_DIM / 32; ++ck) {
        __syncthreads();
#pragma unroll
        for (int i = 0; i < 4; ++i) {
            int idx = tid + i * 256;
            int r = idx >> 3, c4 = (idx & 7) * 4;
            *(float4*)&Xs[r][c4] =
                *(const float4*)(x + (size_t)(rowBase + r) * K_DIM + ck * 32 + c4);
        }
#pragma unroll
        for (int i = 0; i < 3; ++i) {
            int idx = tid + i * 256;
            int r = idx >> 3, c4 = (idx & 7) * 4;
            const float* src = (r < R1) ? (A + (size_t)r * K_DIM)
                                        : (A1 + (size_t)(r - R1) * K_DIM);
            *(float4*)&As[r][c4] = *(const float4*)(src + ck * 32 + c4);
        }
        __syncthreads();

#pragma unroll
        for (int k = 0; k < 32; k += 4) {
            v2f a = *(const v2f*)&Xs[rBase + l16][k + koff];
#pragma unroll
            for (int j = 0; j < 6; ++j) {
                v2f bf = *(const v2f*)&As[j * 16 + l16][k + koff];
                acc[j] = __builtin_amdgcn_wmma_f32_16x16x4_f32(
                    false, a, false, bf, (short)0, acc[j], false, false);
            }
        }
    }

#pragma unroll
    for (int j = 0; j < 6; ++j) {
#pragma unroll
        for (int g = 0; g < 8; ++g) {
            int row = rowBase + rBase + g + ((lane >> 4) ? 8 : 0);
            int cl  = j * 16 + l16;
            float v = acc[j][g];
            if (j < 4) h1[(size_t)row * R1 + cl] = S1 * v;
            else       u[(size_t)row * R2 + (cl - R1)] = v;
        }
    }
}

// ---------------------------------------------------------------------------
// Kernel 2: E2 = 2 * (B1 @ A2)   [32,32]
// ---------------------------------------------------------------------------
__global__ __launch_bounds__(256)
void lora_mid_kernel(const float* __restrict__ A2,
                     const float* __restrict__ B1,
                     float* __restrict__ E2)
{
    for (int e = threadIdx.x; e < R2 * R2; e += 256) {
        int r = e >> 5, p = e & 31;
        float s = 0.f;
#pragma unroll 8
        for (int q = 0; q < R1; ++q)
            s += B1[r * R1 + q] * A2[q * R2 + p];
        E2[e] = 2.0f * s;
    }
}

// ---------------------------------------------------------------------------
// Kernel 3: h4 = u @ E2^T   [M,32]
// ---------------------------------------------------------------------------
__global__ __launch_bounds__(256)
void lora_tail_kernel(const float* __restrict__ u,
                      const float* __restrict__ E2,
                      float* __restrict__ h4)
{
    __shared__ float E2s[R2 * R2];
    for (int e = threadIdx.x; e < R2 * R2; e += 256) E2s[e] = E2[e];
    __syncthreads();

    int g = blockIdx.x * 256 + threadIdx.x;
    int m = g >> 5, r = g & 31;
    float s = 0.f;
#pragma unroll
    for (int p = 0; p < R2; ++p)
        s += u[(size_t)m * R2 + p] * E2s[r * R2 + p];
    h4[(size_t)m * R2 + r] = s;
}

// ---------------------------------------------------------------------------
// Kernel 4: out = x @ W^T + b + h1 @ B^T + h4 @ B2^T
// Effective K = 4096 + 64 + 32 (131 chunks of 32). Double-buffered LDS with
// GLOBAL_LOAD_ASYNC_TO_LDS_B128 prefetch of chunk ck+1 overlapped with the 64
// v_wmma_f32_16x16x4_f32 of chunk ck.
// ---------------------------------------------------------------------------
struct SrcSel { const float* xb; int xld, xc; const float* wb; int wld, wc; };

__device__ __forceinline__ SrcSel pick_chunk(int ck,
    const float* x, const float* W, const float* Bm, const float* B2,
    const float* h1, const float* h4)
{
    SrcSel s;
    if (ck < K_DIM / 32)        { s = { x,  K_DIM, ck * 32, W,  K_DIM, ck * 32 }; }
    else if (ck == K_DIM/32)    { s = { h1, R1,    0,       Bm, R1,    0       }; }
    else if (ck == K_DIM/32+1)  { s = { h1, R1,    32,      Bm, R1,    32      }; }
    else                        { s = { h4, R2,    0,       B2, R2,    0       }; }
    return s;
}

__global__ __launch_bounds__(256)
void fused_gemm_kernel(const float* __restrict__ x,
                       const float* __restrict__ W,
                       const float* __restrict__ bias,
                       const float* __restrict__ Bm,
                       const float* __restrict__ B2,
                       const float* __restrict__ h1,
                       const float* __restrict__ h4,
                       float* __restrict__ out)
{
    __shared__ float Xs[2][128][36];
    __shared__ float Ws[2][128][36];

    const int tid  = threadIdx.x;
    const int wave = tid >> 5;
    const int lane = tid & 31;
    const int l16  = lane & 15;
    const int koff = (lane >> 4) * 2;
    const int wr   = wave & 3;
    const int wc   = wave >> 2;
    const int rowBase = blockIdx.x * 128;
    const int colBase = blockIdx.y * 128;
    const int rB = wr * 32;
    const int cB = wc * 64;
    const int nChunks = K_DIM / 32 + 3;

    v8f acc[2][4];
#pragma unroll
    for (int i = 0; i < 2; ++i)
#pragma unroll
        for (int j = 0; j < 4; ++j) acc[i][j] = (v8f){0,0,0,0,0,0,0,0};

    auto compute = [&](int b) {
#pragma unroll
        for (int k = 0; k < 32; k += 4) {
            v2f a0 = *(const v2f*)&Xs[b][rB + l16][k + koff];
            v2f a1 = *(const v2f*)&Xs[b][rB + 16 + l16][k + koff];
#pragma unroll
            for (int j = 0; j < 4; ++j) {
                v2f bf = *(const v2f*)&Ws[b][cB + j * 16 + l16][k + koff];
                acc[0][j] = __builtin_amdgcn_wmma_f32_16x16x4_f32(
                    false, a0, false, bf, (short)0, acc[0][j], false, false);
                acc[1][j] = __builtin_amdgcn_wmma_f32_16x16x4_f32(
                    false, a1, false, bf, (short)0, acc[1][j], false, false);
            }
        }
    };

#if USE_ASYNC
    // Issue fill of chunk ck into LDS buffer b: 8 async B128 copies per lane.
    auto issue = [&](int ck, int b) {
        SrcSel s = pick_chunk(ck, x, W, Bm, B2, h1, h4);
#pragma unroll
        for (int i = 0; i < 4; ++i) {
            int idx = tid + i * 256;
            int r = idx >> 3, c4 = (idx & 7) * 4;
            async_copy16(s.xb + (size_t)(rowBase + r) * s.xld + s.xc + c4,
                         &Xs[b][r][c4]);
            async_copy16(s.wb + (size_t)(colBase + r) * s.wld + s.wc + c4,
                         &Ws[b][r][c4]);
        }
    };

    issue(0, 0);
    for (int ck = 0; ck < nChunks; ++ck) {
        const int cur = ck & 1;
        if (ck + 1 < nChunks) {
            issue(ck + 1, (ck + 1) & 1);            // prefetch next chunk
            __builtin_amdgcn_s_wait_asynccnt(8);    // wait only for older batch
        } else {
            __builtin_amdgcn_s_wait_asynccnt(0);
        }
        __syncthreads();        // chunk ck visible to all waves
        compute(cur);
        __syncthreads();        // all waves done reading before buffer reuse
    }
#else
    for (int ck = 0; ck < nChunks; ++ck) {
        SrcSel s = pick_chunk(ck, x, W, Bm, B2, h1, h4);
        __syncthreads();
#pragma unroll
        for (int i = 0; i < 4; ++i) {
            int idx = tid + i * 256;
            int r = idx >> 3, c4 = (idx & 7) * 4;
            *(float4*)&Xs[0][r][c4] =
                *(const float4*)(s.xb + (size_t)(rowBase + r) * s.xld + s.xc + c4);
            *(float4*)&Ws[0][r][c4] =
                *(const float4*)(s.wb + (size_t)(colBase + r) * s.wld + s.wc + c4);
        }
        __syncthreads();
        compute(0);
    }
#endif

#pragma unroll
    for (int i = 0; i < 2; ++i) {
#pragma unroll
        for (int j = 0; j < 4; ++j) {
#pragma unroll
            for (int g = 0; g < 8; ++g) {
                int row = rowBase + rB + i * 16 + g + ((lane >> 4) ? 8 : 0);
                int col = colBase + cB + j * 16 + l16;
                out[(size_t)row * N_DIM + col] = acc[i][j][g] + bias[col];
            }
        }
    }
}

// ---------------------------------------------------------------------------
extern "C" void kernel_launch(void* const* d_in, const int* in_sizes, int n_in,
                              void* d_out, int out_size, void* d_ws, size_t ws_size,
                              hipStream_t stream)
{
    const float* x  = (const float*)d_in[0];
    const float* W  = (const float*)d_in[1];
    const float* b  = (const float*)d_in[2];
    const float* A  = (const float*)d_in[3];
    const float* Bm = (const float*)d_in[4];
    const float* A1 = (const float*)d_in[5];
    const float* A2 = (const float*)d_in[6];
    const float* B1 = (const float*)d_in[7];
    const float* B2 = (const float*)d_in[8];
    float* out = (float*)d_out;

    float* h1 = (float*)d_ws;                  // [8192,64]
    float* u  = h1 + (size_t)M_TOT * R1;       // [8192,32]
    float* h4 = u  + (size_t)M_TOT * R2;       // [8192,32]
    float* E2 = h4 + (size_t)M_TOT * R2;       // [32,32]

    lora_proj_kernel<<<M_TOT / 128, 256, 0, stream>>>(x, A, A1, h1, u);
    lora_mid_kernel<<<1, 256, 0, stream>>>(A2, B1, E2);
    lora_tail_kernel<<<(M_TOT * R2) / 256, 256, 0, stream>>>(u, E2, h4);
    fused_gemm_kernel<<<dim3(M_TOT / 128, N_DIM / 128), 256, 0, stream>>>(
        x, W, b, Bm, B2, h1, h4, out);
}